// GATStage4_AttentionSoftmax_51994874085812
// MI455X (gfx1250) — compile-verified
//
#include <hip/hip_runtime.h>

#define TPB 256
#define VEC 4

// Native clang vector types (required by __builtin_nontemporal_load/store).
typedef float     v4f   __attribute__((ext_vector_type(4)));
typedef long long v2i64 __attribute__((ext_vector_type(2)));

// ---------------------------------------------------------------------------
// Pass 0: init node accumulators in workspace.
//   ws layout: [ emax[0..N) | esum[0..N) ]  (floats). N is read from device.
// ---------------------------------------------------------------------------
__global__ void k_init(float* __restrict__ ws, const int* __restrict__ pN) {
    const int N = pN[0];
    float* __restrict__ emax = ws;
    float* __restrict__ esum = ws + N;
    for (int i = blockIdx.x * blockDim.x + threadIdx.x; i < N;
         i += gridDim.x * blockDim.x) {
        emax[i] = -__builtin_inff();
        esum[i] = 0.0f;
    }
}

// ---------------------------------------------------------------------------
// Pass 1: segment max via native GLOBAL_ATOMIC_MAX_NUM_F32 (agent scope).
// Edge streams use non-temporal hints so the hot node arrays stay in L2.
// ---------------------------------------------------------------------------
__global__ void k_segmax(const float* __restrict__ e,
                         const long long* __restrict__ tgt,
                         float* __restrict__ ws,
                         long long E) {
    float* __restrict__ emax = ws;
    long long i0 = (long long)VEC * ((long long)blockIdx.x * blockDim.x + threadIdx.x);
    if (i0 + (VEC - 1) < E) {
        v4f   ev  = __builtin_nontemporal_load((const v4f*)(e + i0));
        v2i64 t01 = __builtin_nontemporal_load((const v2i64*)(tgt + i0));
        v2i64 t23 = __builtin_nontemporal_load((const v2i64*)(tgt + i0 + 2));
        __hip_atomic_fetch_max(&emax[(int)t01.x], ev.x, __ATOMIC_RELAXED, __HIP_MEMORY_SCOPE_AGENT);
        __hip_atomic_fetch_max(&emax[(int)t01.y], ev.y, __ATOMIC_RELAXED, __HIP_MEMORY_SCOPE_AGENT);
        __hip_atomic_fetch_max(&emax[(int)t23.x], ev.z, __ATOMIC_RELAXED, __HIP_MEMORY_SCOPE_AGENT);
        __hip_atomic_fetch_max(&emax[(int)t23.y], ev.w, __ATOMIC_RELAXED, __HIP_MEMORY_SCOPE_AGENT);
    } else {
        for (long long i = i0; i < E && i < i0 + VEC; ++i) {
            float v = e[i];
            int   t = (int)tgt[i];
            __hip_atomic_fetch_max(&emax[t], v, __ATOMIC_RELAXED, __HIP_MEMORY_SCOPE_AGENT);
        }
    }
}

// ---------------------------------------------------------------------------
// Pass 2: exp_e = exp(e - emax[tgt]); segment sum via GLOBAL_ATOMIC_ADD_F32.
// ---------------------------------------------------------------------------
__global__ void k_segsum(const float* __restrict__ e,
                         const long long* __restrict__ tgt,
                         float* __restrict__ ws,
                         const int* __restrict__ pN,
                         long long E) {
    const int N = pN[0];
    const float* __restrict__ emax = ws;
    float* __restrict__ esum = ws + N;
    long long i0 = (long long)VEC * ((long long)blockIdx.x * blockDim.x + threadIdx.x);
    if (i0 + (VEC - 1) < E) {
        v4f   ev  = __builtin_nontemporal_load((const v4f*)(e + i0));
        v2i64 t01 = __builtin_nontemporal_load((const v2i64*)(tgt + i0));
        v2i64 t23 = __builtin_nontemporal_load((const v2i64*)(tgt + i0 + 2));
        int ta = (int)t01.x, tb = (int)t01.y, tc = (int)t23.x, td = (int)t23.y;
        float xa = expf(ev.x - emax[ta]);
        float xb = expf(ev.y - emax[tb]);
        float xc = expf(ev.z - emax[tc]);
        float xd = expf(ev.w - emax[td]);
        __hip_atomic_fetch_add(&esum[ta], xa, __ATOMIC_RELAXED, __HIP_MEMORY_SCOPE_AGENT);
        __hip_atomic_fetch_add(&esum[tb], xb, __ATOMIC_RELAXED, __HIP_MEMORY_SCOPE_AGENT);
        __hip_atomic_fetch_add(&esum[tc], xc, __ATOMIC_RELAXED, __HIP_MEMORY_SCOPE_AGENT);
        __hip_atomic_fetch_add(&esum[td], xd, __ATOMIC_RELAXED, __HIP_MEMORY_SCOPE_AGENT);
    } else {
        for (long long i = i0; i < E && i < i0 + VEC; ++i) {
            int t = (int)tgt[i];
            float x = expf(e[i] - emax[t]);
            __hip_atomic_fetch_add(&esum[t], x, __ATOMIC_RELAXED, __HIP_MEMORY_SCOPE_AGENT);
        }
    }
}

// ---------------------------------------------------------------------------
// Pass 3: alpha = exp(e - emax[tgt]) / (esum[tgt] + 1e-16); NT stores.
// ---------------------------------------------------------------------------
__global__ void k_norm(const float* __restrict__ e,
                       const long long* __restrict__ tgt,
                       const float* __restrict__ ws,
                       const int* __restrict__ pN,
                       float* __restrict__ out,
                       long long E) {
    const int N = pN[0];
    const float* __restrict__ emax = ws;
    const float* __restrict__ esum = ws + N;
    long long i0 = (long long)VEC * ((long long)blockIdx.x * blockDim.x + threadIdx.x);
    if (i0 + (VEC - 1) < E) {
        v4f   ev  = __builtin_nontemporal_load((const v4f*)(e + i0));
        v2i64 t01 = __builtin_nontemporal_load((const v2i64*)(tgt + i0));
        v2i64 t23 = __builtin_nontemporal_load((const v2i64*)(tgt + i0 + 2));
        int ta = (int)t01.x, tb = (int)t01.y, tc = (int)t23.x, td = (int)t23.y;
        v4f a;
        a.x = expf(ev.x - emax[ta]) / (esum[ta] + 1e-16f);
        a.y = expf(ev.y - emax[tb]) / (esum[tb] + 1e-16f);
        a.z = expf(ev.z - emax[tc]) / (esum[tc] + 1e-16f);
        a.w = expf(ev.w - emax[td]) / (esum[td] + 1e-16f);
        __builtin_nontemporal_store(a, (v4f*)(out + i0));
    } else {
        for (long long i = i0; i < E && i < i0 + VEC; ++i) {
            int t = (int)tgt[i];
            out[i] = expf(e[i] - emax[t]) / (esum[t] + 1e-16f);
        }
    }
}

extern "C" void kernel_launch(void* const* d_in, const int* in_sizes, int n_in,
                              void* d_out, int out_size, void* d_ws, size_t ws_size,
                              hipStream_t stream) {
    const float*     e          = (const float*)d_in[0];
    const long long* edge_index = (const long long*)d_in[1];
    const int*       pN         = (const int*)d_in[2];
    const long long  E          = (long long)in_sizes[0];
    const long long* tgt        = edge_index + E;   // row 1 of [2, E]
    float*           ws         = (float*)d_ws;     // [emax | esum]
    float*           out        = (float*)d_out;

    const int edgeBlocks = (int)((E + (long long)(VEC * TPB) - 1) / (long long)(VEC * TPB));

    k_init  <<<256,        TPB, 0, stream>>>(ws, pN);
    k_segmax<<<edgeBlocks, TPB, 0, stream>>>(e, tgt, ws, E);
    k_segsum<<<edgeBlocks, TPB, 0, stream>>>(e, tgt, ws, pN, E);
    k_norm  <<<edgeBlocks, TPB, 0, stream>>>(e, tgt, ws, pN, out, E);
}